// HierGatChemEncoder_7009386627250
// MI455X (gfx1250) — compile-verified
//
#include <hip/hip_runtime.h>
#include <math.h>

#define B_  8
#define S_  512
#define E_  32768
#define ND_ 160
#define ED_ 64
#define D_  256
#define FF_ 1024
#define L_  4
#define H_  8
#define DH_ 32
#define NEG_ -1000000000.0f

#define KCHUNK  32
#define BSTRIDE 36   // floats; 36*4=144 bytes -> every row 8-byte aligned

typedef __attribute__((ext_vector_type(2))) float v2f;
typedef __attribute__((ext_vector_type(8))) float v8f;

#if __has_builtin(__builtin_amdgcn_global_load_async_to_lds_b32) && \
    __has_builtin(__builtin_amdgcn_s_wait_asynccnt)
#define HAVE_ASYNC 1
typedef __attribute__((address_space(1))) int as1_int;  // global
typedef __attribute__((address_space(3))) int as3_int;  // LDS
#else
#define HAVE_ASYNC 0
#endif

// float atomic max via sign-split integer atomics (valid for mixed signs when
// the location is initialized to a negative value, which it is: NEG_).
__device__ inline void atomicMaxF(float* addr, float val) {
  if (val >= 0.0f) atomicMax((int*)addr, __float_as_int(val));
  else             atomicMin((unsigned int*)addr, __float_as_uint(val));
}

// ---------------------------------------------------------------------------
// Generic GEMM: C[M,N] = A[M,K] @ Bm[K,N] + bias[N] (optional ReLU).
// Row-major, fully tiled: M % 128 == 0, N % 16 == 0, K % 32 == 0.
// Block = 256 threads = 8 waves; wave w computes 16x16 tile at row (by*8+w)*16,
// cols n0..n0+15 using V_WMMA_F32_16X16X4_F32 (full f32, matches reference).
// The shared K x 16 weight tile is staged into LDS (async global->LDS when the
// gfx1250 builtins exist) transposed so each lane's B fragment is a ds_load_b64.
// A fragments (two adjacent K elements) are single global_load_b64.
// ---------------------------------------------------------------------------
__global__ __launch_bounds__(256)
void gemm_bias_kernel(const float* __restrict__ A,
                      const float* __restrict__ Bm,
                      const float* __restrict__ bias,
                      float* __restrict__ C,
                      int M, int N, int K, int relu)
{
  __shared__ float Bs[16 * BSTRIDE];  // Bs[c*BSTRIDE + kr] = B[k0+kr][n0+c]

  const int tid  = threadIdx.x;
  const int lane = tid & 31;
  const int wave = tid >> 5;
  const int n0 = blockIdx.x * 16;
  const int m0 = (blockIdx.y * 8 + wave) * 16;
  const int lm   = lane & 15;
  const int koff = (lane < 16) ? 0 : 2;

  const float* Arow = A + (size_t)(m0 + lm) * K;
  v8f acc = {};

  for (int k0 = 0; k0 < K; k0 += KCHUNK) {
    // ---- stage B[k0..k0+32)[n0..n0+16) into LDS, transposed ----
#pragma unroll
    for (int i = 0; i < 2; ++i) {
      const int idx = tid + i * 256;      // 0..511
      const int c  = idx & 15;            // column (fast -> coalesced global)
      const int kr = idx >> 4;            // k row 0..31
      const float* gp = &Bm[(size_t)(k0 + kr) * N + n0 + c];
      float* lp = &Bs[c * BSTRIDE + kr];
#if HAVE_ASYNC
      __builtin_amdgcn_global_load_async_to_lds_b32(
          (as1_int*)(void*)gp, (as3_int*)(void*)lp, 0, 0);
#else
      *lp = *gp;
#endif
    }
    if (k0 + KCHUNK < K)                  // speculative prefetch of next A chunk
      __builtin_prefetch(Arow + k0 + KCHUNK, 0, 1);
#if HAVE_ASYNC
    __builtin_amdgcn_s_wait_asynccnt(0);
#endif
    __syncthreads();

    // ---- 8 WMMAs per chunk: A via b64 global loads, B via b64 LDS loads ----
#pragma unroll
    for (int kk = 0; kk < KCHUNK; kk += 4) {
      v2f a = *(const v2f*)(Arow + k0 + kk + koff);
      v2f b = *(const v2f*)(&Bs[lm * BSTRIDE + kk + koff]);
      acc = __builtin_amdgcn_wmma_f32_16x16x4_f32(
          /*neg_a=*/false, a, /*neg_b=*/false, b,
          /*c_mod=*/(short)0, acc, /*reuse_a=*/false, /*reuse_b=*/false);
    }
    __syncthreads();
  }

  const int col = n0 + lm;
  const float bval = bias ? bias[col] : 0.0f;
  const int rbase = m0 + ((lane < 16) ? 0 : 8);
#pragma unroll
  for (int r = 0; r < 8; ++r) {
    float o = acc[r] + bval;
    if (relu) o = fmaxf(o, 0.0f);
    C[(size_t)(rbase + r) * N + col] = o;
  }
}

// Initialize per-layer attention accumulators.
__global__ void init_attn_kernel(float* __restrict__ mbuf,
                                 float* __restrict__ den,
                                 float* __restrict__ agg)
{
  int i = blockIdx.x * blockDim.x + threadIdx.x;
  if (i < B_ * S_ * H_) { mbuf[i] = NEG_; den[i] = 0.0f; }
  if (i < B_ * S_ * D_) agg[i] = 0.0f;
}

// Per (edge, head): logit = scale * q[dst]·(k[src]+e), masked; scatter-max.
__global__ void logits_max_kernel(const int* __restrict__ ei,
                                  const unsigned char* __restrict__ emask,
                                  const float* __restrict__ q,
                                  const float* __restrict__ k,
                                  const float* __restrict__ ebuf,
                                  float* __restrict__ logits,
                                  float* __restrict__ mbuf,
                                  int b)
{
  int t = blockIdx.x * blockDim.x + threadIdx.x;
  if (t >= E_ * H_) return;
  int e = t >> 3;
  int h = t & 7;
  int src = ei[(size_t)b * 2 * E_ + e];
  int dst = ei[(size_t)b * 2 * E_ + E_ + e];
  const float* qp = q + (size_t)(b * S_ + dst) * D_ + h * DH_;
  const float* kp = k + (size_t)(b * S_ + src) * D_ + h * DH_;
  const float* ep = ebuf + (size_t)e * D_ + h * DH_;
  float acc = 0.0f;
#pragma unroll
  for (int d = 0; d < DH_; ++d) acc = fmaf(qp[d], kp[d] + ep[d], acc);
  float lg = acc * 0.17677669529663688f;  // 1/sqrt(Dh), TEMP=1
  if (!emask[(size_t)b * E_ + e]) lg = NEG_;
  logits[t] = lg;
  atomicMaxF(&mbuf[(size_t)(b * S_ + dst) * H_ + h], lg);
}

// Per (edge, head): ex = exp(l - m[seg]); den += ex; agg += ex * (v[src]+e).
// (Normalization by den is deferred — mathematically identical to reference.)
__global__ void expsum_kernel(const int* __restrict__ ei,
                              const float* __restrict__ logits,
                              const float* __restrict__ mbuf,
                              const float* __restrict__ v,
                              const float* __restrict__ ebuf,
                              float* __restrict__ den,
                              float* __restrict__ agg,
                              int b)
{
  int t = blockIdx.x * blockDim.x + threadIdx.x;
  if (t >= E_ * H_) return;
  int e = t >> 3;
  int h = t & 7;
  int src = ei[(size_t)b * 2 * E_ + e];
  int dst = ei[(size_t)b * 2 * E_ + E_ + e];
  int seg = b * S_ + dst;
  float ex = expf(logits[t] - mbuf[(size_t)seg * H_ + h]);
  atomicAdd(&den[(size_t)seg * H_ + h], ex);
  const float* vp = v + (size_t)(b * S_ + src) * D_ + h * DH_;
  const float* ep = ebuf + (size_t)e * D_ + h * DH_;
  float* ap = agg + (size_t)seg * D_ + h * DH_;
#pragma unroll
  for (int d = 0; d < DH_; ++d) atomicAdd(&ap[d], ex * (vp[d] + ep[d]));
}

// agg[node, h*32+d] /= (den[node,h] + 1e-9)
__global__ void attn_norm_kernel(float* __restrict__ agg,
                                 const float* __restrict__ den)
{
  int i = blockIdx.x * blockDim.x + threadIdx.x;
  if (i >= B_ * S_ * D_) return;
  int node = i / D_;
  int h = (i % D_) / DH_;
  agg[i] = agg[i] / (den[(size_t)node * H_ + h] + 1e-9f);
}

// out = LN(hin + add1 + add2) * scale + bias, optionally zeroed by node mask.
// One block (D_ threads) per row; biased variance (matches jnp.var), eps 1e-5.
__global__ void add_ln_kernel(const float* __restrict__ hin,
                              const float* __restrict__ add1,
                              const float* __restrict__ add2,
                              const float* __restrict__ sc,
                              const float* __restrict__ bi,
                              const unsigned char* __restrict__ mask,
                              float* __restrict__ hout)
{
  const int row = blockIdx.x;
  const int d = threadIdx.x;
  const size_t idx = (size_t)row * D_ + d;
  float v = hin[idx];
  if (add1) v += add1[idx];
  if (add2) v += add2[idx];
  __shared__ float rs[D_];
  __shared__ float rq[D_];
  rs[d] = v;
  rq[d] = v * v;
  __syncthreads();
  for (int off = D_ / 2; off > 0; off >>= 1) {
    if (d < off) { rs[d] += rs[d + off]; rq[d] += rq[d + off]; }
    __syncthreads();
  }
  const float mu  = rs[0] * (1.0f / D_);
  const float var = rq[0] * (1.0f / D_) - mu * mu;
  float o = (v - mu) * rsqrtf(var + 1e-5f) * sc[d] + bi[d];
  if (mask && !mask[row]) o = 0.0f;
  hout[idx] = o;
}

extern "C" void kernel_launch(void* const* d_in, const int* in_sizes, int n_in,
                              void* d_out, int out_size, void* d_ws, size_t ws_size,
                              hipStream_t stream)
{
  (void)in_sizes; (void)n_in; (void)out_size; (void)ws_size;
  const float* x    = (const float*)d_in[0];
  const int*   ei   = (const int*)d_in[1];
  const float* ea   = (const float*)d_in[2];
  const unsigned char* nmask = (const unsigned char*)d_in[3];
  const unsigned char* emask = (const unsigned char*)d_in[4];
  const float* Wp = (const float*)d_in[5];
  const float* bp = (const float*)d_in[6];
  const float* Wq = (const float*)d_in[7];
  const float* bq = (const float*)d_in[8];
  const float* Wk = (const float*)d_in[9];
  const float* bk = (const float*)d_in[10];
  const float* Wv = (const float*)d_in[11];
  const float* bv = (const float*)d_in[12];
  const float* We = (const float*)d_in[13];
  const float* be = (const float*)d_in[14];
  const float* Wo = (const float*)d_in[15];
  const float* bo = (const float*)d_in[16];
  const float* ln1s = (const float*)d_in[17];
  const float* ln1b = (const float*)d_in[18];
  const float* W1 = (const float*)d_in[19];
  const float* b1 = (const float*)d_in[20];
  const float* W2 = (const float*)d_in[21];
  const float* b2 = (const float*)d_in[22];
  const float* ln2s = (const float*)d_in[23];
  const float* ln2b = (const float*)d_in[24];
  const float* ns = (const float*)d_in[25];
  const float* nb = (const float*)d_in[26];

  float* ws = (float*)d_ws;
  size_t o = 0;
  const size_t BS = (size_t)B_ * S_;
  float* xp   = ws + o; o += BS * D_;
  float* h    = ws + o; o += BS * D_;
  float* q    = ws + o; o += BS * D_;
  float* k    = ws + o; o += BS * D_;
  float* v    = ws + o; o += BS * D_;
  float* ebuf = ws + o; o += (size_t)E_ * D_;    // per-batch edge features
  float* logits = ws + o; o += (size_t)E_ * H_;  // per-batch logits
  float* mbuf = ws + o; o += BS * H_;
  float* den  = ws + o; o += BS * H_;
  float* agg  = ws + o; o += BS * D_;
  float* attn = ws + o; o += BS * D_;
  float* ff1  = ws + o; o += BS * FF_;
  float* ff2  = ws + o; o += BS * D_;
  (void)o;

  auto gemm = [&](const float* A, const float* Bm, const float* bias, float* C,
                  int M, int N, int K, int relu) {
    dim3 grid(N / 16, M / 128);
    gemm_bias_kernel<<<grid, dim3(256), 0, stream>>>(A, Bm, bias, C, M, N, K, relu);
  };

  // xp = x @ Wp + bp ; h = xp
  gemm(x, Wp, bp, xp, B_ * S_, D_, ND_, 0);
  (void)hipMemcpyAsync(h, xp, BS * D_ * sizeof(float), hipMemcpyDeviceToDevice,
                       stream);

  const int EH = E_ * H_;
  for (int l = 0; l < L_; ++l) {
    gemm(h, Wq + (size_t)l * D_ * D_, bq + (size_t)l * D_, q, B_ * S_, D_, D_, 0);
    gemm(h, Wk + (size_t)l * D_ * D_, bk + (size_t)l * D_, k, B_ * S_, D_, D_, 0);
    gemm(h, Wv + (size_t)l * D_ * D_, bv + (size_t)l * D_, v, B_ * S_, D_, D_, 0);

    init_attn_kernel<<<(B_ * S_ * D_ + 255) / 256, 256, 0, stream>>>(mbuf, den, agg);

    for (int b = 0; b < B_; ++b) {
      // e = edge_attr[b] @ We[l] + be[l]   (E x 64) @ (64 x 256)
      gemm(ea + (size_t)b * E_ * ED_, We + (size_t)l * ED_ * D_,
           be + (size_t)l * D_, ebuf, E_, D_, ED_, 0);
      logits_max_kernel<<<(EH + 255) / 256, 256, 0, stream>>>(
          ei, emask, q, k, ebuf, logits, mbuf, b);
      expsum_kernel<<<(EH + 255) / 256, 256, 0, stream>>>(
          ei, logits, mbuf, v, ebuf, den, agg, b);
    }

    attn_norm_kernel<<<(B_ * S_ * D_ + 255) / 256, 256, 0, stream>>>(agg, den);
    gemm(agg, Wo + (size_t)l * D_ * D_, bo + (size_t)l * D_, attn, B_ * S_, D_, D_, 0);
    // h = LN(h + attn + xp)
    add_ln_kernel<<<B_ * S_, D_, 0, stream>>>(h, attn, xp,
        ln1s + (size_t)l * D_, ln1b + (size_t)l * D_, nullptr, h);
    // FFN
    gemm(h, W1 + (size_t)l * D_ * FF_, b1 + (size_t)l * FF_, ff1, B_ * S_, FF_, D_, 1);
    gemm(ff1, W2 + (size_t)l * FF_ * D_, b2 + (size_t)l * D_, ff2, B_ * S_, D_, FF_, 0);
    // h = LN(h + ff) * node_mask
    add_ln_kernel<<<B_ * S_, D_, 0, stream>>>(h, ff2, nullptr,
        ln2s + (size_t)l * D_, ln2b + (size_t)l * D_, nmask, h);
  }

  // out = LN(h, norm_s, norm_b)
  add_ln_kernel<<<B_ * S_, D_, 0, stream>>>(h, nullptr, nullptr, ns, nb, nullptr,
                                            (float*)d_out);
}